// Mamba_29523605193107
// MI455X (gfx1250) — compile-verified
//
#include <hip/hip_runtime.h>
#include <hip/hip_bf16.h>
#include <math.h>

// ---------------------------------------------------------------------------
// Mamba 6-branch block for MI455X (gfx1250, wave32, WMMA f32 16x16x4)
// Two schedules chosen from ws_size (deterministic):
//  - BATCHED  (~172.5 MB ws): all 6 branches concurrent via blockIdx.z;
//    scans write per-branch buffers, reduce kernel sums (no atomics).
//  - SEQUENTIAL (~42.7 MB ws): branch buffers reused, scans += into total.
// ---------------------------------------------------------------------------
typedef float v2f __attribute__((ext_vector_type(2)));
typedef float v8f __attribute__((ext_vector_type(8)));

#define D_MODEL 256
#define D_INNER 512
#define D_STATE 16
#define DT_RANK 16
#define R2N     48
#define SEQLEN  2048
#define BATCH   2
#define TCH     64

#define SZ_XZ    (BATCH * 2 * D_INNER * SEQLEN)   // 4,194,304
#define SZ_TOT   (BATCH * D_INNER * SEQLEN)       // 2,097,152
#define SZ_U     (BATCH * D_INNER * SEQLEN)       // 2,097,152
#define SZ_XDBL  (BATCH * R2N * SEQLEN)           //   196,608
#define SZ_DELTA (BATCH * D_INNER * SEQLEN)       // 2,097,152

__device__ __forceinline__ float siluf(float x) { return x / (1.0f + __expf(-x)); }
__device__ __forceinline__ float softplusf(float x) {
  return (x > 20.0f) ? x : log1pf(__expf(x));
}

// scan position t -> original l for reading xz (input permutation)
__device__ __forceinline__ int perm_in(int br, int t) {
  switch (br) {
    case 0: return t;
    case 1: return (SEQLEN - 1) - t;
    case 2: return (t & 63) * 32 + (t >> 6);
    case 3: { int s = (SEQLEN - 1) - t; return (s & 63) * 32 + (s >> 6); }
    case 4: return (t & 15) * 128 + (t >> 4);
    default:{ int s = (SEQLEN - 1) - t; return (s & 15) * 128 + (s >> 4); }
  }
}
// scan position t -> original l for output (reference does not un-flip 3/5)
__device__ __forceinline__ int perm_out(int br, int t) {
  switch (br) {
    case 0: return t;
    case 1: return (SEQLEN - 1) - t;
    case 2: case 3: return (t & 63) * 32 + (t >> 6);
    default: return (t & 15) * 128 + (t >> 4);
  }
}

// ---------------------------------------------------------------------------
__global__ void k_zero(float* __restrict__ p, int n) {
  int i = blockIdx.x * blockDim.x + threadIdx.x;
  if (i < n) p[i] = 0.0f;
}

// total[i] = sum_br ybr[br][i]
__global__ void k_reduce(const float* __restrict__ ybr, float* __restrict__ total, int n) {
  int i = blockIdx.x * blockDim.x + threadIdx.x;
  if (i >= n) return;
  float s = 0.0f;
#pragma unroll
  for (int br = 0; br < 6; ++br) s += ybr[(size_t)br * n + i];
  total[i] = s;
}

// ---------------------------------------------------------------------------
// xz[b,e,l] = sum_d in_w[e,d]*hs[b,l,d]  (M=e=1024, N=bl=4096, K=256)
// One wave -> one 16(M) x 32(N) slab: shared A fragment, two accumulators.
__global__ __launch_bounds__(32) void k_in_proj(const float* __restrict__ hs,
                                                const float* __restrict__ w,
                                                float* __restrict__ xz) {
  const int lane = threadIdx.x;
  const int m0 = blockIdx.x << 4;           // e tile
  const int n0 = blockIdx.y << 5;           // bl tile pair
  const int rr = lane & 15;
  const int kh = (lane >> 4) << 1;
  const float* aptr  = w  + (m0 + rr) * D_MODEL + kh;
  const float* bptr0 = hs + (n0 + rr) * D_MODEL + kh;
  const float* bptr1 = bptr0 + 16 * D_MODEL;
  v8f acc0 = {}, acc1 = {};
  for (int k0 = 0; k0 < D_MODEL; k0 += 4) {
    __builtin_prefetch(aptr + k0 + 64, 0, 1);       // global_prefetch_b8
    v2f a  = *(const v2f*)(aptr + k0);
    v2f b0 = *(const v2f*)(bptr0 + k0);
    v2f b1 = *(const v2f*)(bptr1 + k0);
    acc0 = __builtin_amdgcn_wmma_f32_16x16x4_f32(false, a, false, b0, (short)0, acc0, false, false);
    acc1 = __builtin_amdgcn_wmma_f32_16x16x4_f32(false, a, false, b1, (short)0, acc1, false, false);
  }
  const int mb = (lane >> 4) << 3;
  const int bl0 = n0 + rr;                  // n0 is 32-aligned -> same b for both tiles
  const size_t base = (size_t)(bl0 >> 11) * (2 * D_INNER) * SEQLEN + (bl0 & 2047);
#pragma unroll
  for (int v = 0; v < 8; ++v) {
    const size_t row = (size_t)(m0 + mb + v) * SEQLEN;
    xz[base + row]      = acc0[v];
    xz[base + row + 16] = acc1[v];
  }
}

// ---------------------------------------------------------------------------
// u[b,d,t] = silu(conv_b[d] + sum_k conv_w[d,k]*xz[b,d,perm_in(t-3+k)])
__global__ void k_conv_silu(const float* __restrict__ xz,
                            const float* __restrict__ cw,
                            const float* __restrict__ cb,
                            float* __restrict__ u, size_t ustride, int br0) {
  const int br = br0 + blockIdx.z;
  float* ub = u + (size_t)blockIdx.z * ustride;
  int idx = blockIdx.x * blockDim.x + threadIdx.x;
  if (idx >= BATCH * D_INNER * SEQLEN) return;
  int t = idx & (SEQLEN - 1);
  int d = (idx >> 11) & (D_INNER - 1);
  int b = idx >> 20;
  const float* xzb = xz + (size_t)b * (2 * D_INNER) * SEQLEN + (size_t)d * SEQLEN;
  const float* wk  = cw + (br * D_INNER + d) * 4;
  float s = cb[br * D_INNER + d];
#pragma unroll
  for (int k = 0; k < 4; ++k) {
    int tt = t - 3 + k;
    if (tt >= 0) s += wk[k] * xzb[perm_in(br, tt)];
  }
  ub[idx] = siluf(s);
}

// ---------------------------------------------------------------------------
// xdbl[b,r,t] = sum_d xproj_w[br][r,d]*u[b,d,t]  (M=48, N=4096, K=512)
__global__ __launch_bounds__(32) void k_xproj(const float* __restrict__ u, size_t ustride,
                                              const float* __restrict__ xw,
                                              float* __restrict__ xdbl, size_t xstride,
                                              int br0) {
  const int br = br0 + blockIdx.z;
  const float* ub = u + (size_t)blockIdx.z * ustride;
  float* xb = xdbl + (size_t)blockIdx.z * xstride;
  const int lane = threadIdx.x;
  const int m0 = blockIdx.x << 4;           // r tile (0,16,32)
  const int n0 = blockIdx.y << 5;           // bt tile pair
  const int rr = lane & 15;
  const int kh = (lane >> 4) << 1;
  const int b = n0 >> 11, t = n0 & 2047;
  const float* A  = xw + (size_t)br * R2N * D_INNER + (m0 + rr) * D_INNER + kh;
  const float* B0 = ub + (size_t)b * D_INNER * SEQLEN + (size_t)kh * SEQLEN + (t + rr);
  v8f acc0 = {}, acc1 = {};
  for (int k0 = 0; k0 < D_INNER; k0 += 4) {
    v2f a = *(const v2f*)(A + k0);
    const float* p = B0 + (size_t)k0 * SEQLEN;
    v2f b0, b1;
    b0.x = p[0];      b0.y = p[SEQLEN];
    b1.x = p[16];     b1.y = p[SEQLEN + 16];
    acc0 = __builtin_amdgcn_wmma_f32_16x16x4_f32(false, a, false, b0, (short)0, acc0, false, false);
    acc1 = __builtin_amdgcn_wmma_f32_16x16x4_f32(false, a, false, b1, (short)0, acc1, false, false);
  }
  const int mb = (lane >> 4) << 3;
#pragma unroll
  for (int v = 0; v < 8; ++v) {
    const size_t o = (size_t)b * R2N * SEQLEN + (size_t)(m0 + mb + v) * SEQLEN + (t + rr);
    xb[o]      = acc0[v];
    xb[o + 16] = acc1[v];
  }
}

// ---------------------------------------------------------------------------
// delta[b,d,t] = softplus(sum_r dt_w[br][d,r]*xdbl[b,r,t] + dt_b[br][d])
__global__ __launch_bounds__(32) void k_dtproj(const float* __restrict__ xdbl, size_t xstride,
                                               const float* __restrict__ dw,
                                               const float* __restrict__ db,
                                               float* __restrict__ delta, size_t dstride,
                                               int br0) {
  const int br = br0 + blockIdx.z;
  const float* xb = xdbl + (size_t)blockIdx.z * xstride;
  float* de = delta + (size_t)blockIdx.z * dstride;
  const int lane = threadIdx.x;
  const int m0 = blockIdx.x << 4;           // d tile (32)
  const int n0 = blockIdx.y << 5;           // bt tile pair
  const int rr = lane & 15;
  const int kh = (lane >> 4) << 1;
  const int b = n0 >> 11, t = n0 & 2047;
  const float* A  = dw + (size_t)br * D_INNER * DT_RANK + (m0 + rr) * DT_RANK + kh;
  const float* B0 = xb + (size_t)b * R2N * SEQLEN + (size_t)kh * SEQLEN + (t + rr);
  v8f acc0 = {}, acc1 = {};
#pragma unroll
  for (int k0 = 0; k0 < DT_RANK; k0 += 4) {
    v2f a = *(const v2f*)(A + k0);
    const float* p = B0 + (size_t)k0 * SEQLEN;
    v2f b0, b1;
    b0.x = p[0];      b0.y = p[SEQLEN];
    b1.x = p[16];     b1.y = p[SEQLEN + 16];
    acc0 = __builtin_amdgcn_wmma_f32_16x16x4_f32(false, a, false, b0, (short)0, acc0, false, false);
    acc1 = __builtin_amdgcn_wmma_f32_16x16x4_f32(false, a, false, b1, (short)0, acc1, false, false);
  }
  const int mb = (lane >> 4) << 3;
#pragma unroll
  for (int v = 0; v < 8; ++v) {
    const int d = m0 + mb + v;
    const float bias = db[br * D_INNER + d];
    const size_t o = (size_t)b * D_INNER * SEQLEN + (size_t)d * SEQLEN + (t + rr);
    de[o]      = softplusf(acc0[v] + bias);
    de[o + 16] = softplusf(acc1[v] + bias);
  }
}

// ---------------------------------------------------------------------------
// Selective scan: one (b,d) chain per lane, 16 states in registers,
// B/C vectors staged in LDS per 64-step chunk.
// accum==0: dst[b,d,perm_out(t)]  = gated y   (per-branch buffer, batched)
// accum==1: dst[b,d,perm_out(t)] += gated y   (shared total, serialized)
__global__ __launch_bounds__(128) void k_scan(const float* __restrict__ u, size_t ustride,
                                              const float* __restrict__ delta, size_t dstride,
                                              const float* __restrict__ xdbl, size_t xstride,
                                              const float* __restrict__ xz,
                                              const float* __restrict__ A_log,
                                              const float* __restrict__ D_skip,
                                              float* __restrict__ dst, size_t ystride,
                                              int br0, int accum) {
  __shared__ float sB[TCH][D_STATE];
  __shared__ float sC[TCH][D_STATE];
  const int br = br0 + blockIdx.z;
  const int b = blockIdx.x >> 2;
  const int d = ((blockIdx.x & 3) << 7) + threadIdx.x;

  float Areg[D_STATE], h[D_STATE];
#pragma unroll
  for (int n = 0; n < D_STATE; ++n) {
    Areg[n] = -__expf(A_log[((size_t)br * D_INNER + d) * D_STATE + n]);
    h[n] = 0.0f;
  }
  const float dsk = D_skip[br * D_INNER + d];
  const float* up = u + (size_t)blockIdx.z * ustride
                      + (size_t)b * D_INNER * SEQLEN + (size_t)d * SEQLEN;
  const float* dp = delta + (size_t)blockIdx.z * dstride
                          + (size_t)b * D_INNER * SEQLEN + (size_t)d * SEQLEN;
  const float* bc = xdbl + (size_t)blockIdx.z * xstride
                         + (size_t)b * R2N * SEQLEN + (size_t)DT_RANK * SEQLEN;
  const float* zp = xz + (size_t)b * (2 * D_INNER) * SEQLEN
                       + (size_t)(D_INNER + d) * SEQLEN;
  float* tp = dst + (size_t)blockIdx.z * ystride
                  + (size_t)b * D_INNER * SEQLEN + (size_t)d * SEQLEN;

  for (int c0 = 0; c0 < SEQLEN; c0 += TCH) {
    __syncthreads();
    for (int j = threadIdx.x; j < 32 * TCH; j += blockDim.x) {
      int n  = j / TCH;
      int tc = j & (TCH - 1);
      float v = bc[(size_t)n * SEQLEN + (c0 + tc)];
      if (n < D_STATE) sB[tc][n] = v; else sC[tc][n - D_STATE] = v;
    }
    __syncthreads();
#pragma unroll 4
    for (int tc = 0; tc < TCH; ++tc) {
      const int t = c0 + tc;
      const float dl  = dp[t];
      const float uv  = up[t];
      const float dbu = dl * uv;
      float y = 0.0f;
#pragma unroll
      for (int n = 0; n < D_STATE; ++n) {
        float dA = __expf(dl * Areg[n]);      // v_exp_f32 (trans pipe)
        h[n] = dA * h[n] + dbu * sB[tc][n];
        y = fmaf(h[n], sC[tc][n], y);
      }
      y = fmaf(uv, dsk, y);
      const float zv = zp[perm_in(br, t)];
      const float val = y * siluf(zv);
      const int lo = perm_out(br, t);
      if (accum) tp[lo] += val;               // serialized across branches
      else       tp[lo]  = val;               // private per-branch buffer
    }
  }
}

// ---------------------------------------------------------------------------
// out[b,l,o] = sum_d out_w[o,d]*total[b,d,l]  (M=bl=4096, N=o=256, K=512)
// Shared A fragment across two o-tiles per wave.
__global__ __launch_bounds__(32) void k_out_proj(const float* __restrict__ total,
                                                 const float* __restrict__ ow,
                                                 float* __restrict__ out) {
  const int lane = threadIdx.x;
  const int m0 = blockIdx.x << 4;           // bl tile
  const int n0 = blockIdx.y << 5;           // o tile pair
  const int rr = lane & 15;
  const int kh = (lane >> 4) << 1;
  const int bl = m0 + rr;
  const int b = bl >> 11, l = bl & 2047;
  const float* A  = total + (size_t)b * D_INNER * SEQLEN + (size_t)kh * SEQLEN + l;
  const float* B0 = ow + (n0 + rr) * D_INNER + kh;
  const float* B1 = B0 + 16 * D_INNER;
  v8f acc0 = {}, acc1 = {};
  for (int k0 = 0; k0 < D_INNER; k0 += 4) {
    const float* p = A + (size_t)k0 * SEQLEN;
    v2f a;
    a.x = p[0];
    a.y = p[SEQLEN];
    v2f b0 = *(const v2f*)(B0 + k0);
    v2f b1 = *(const v2f*)(B1 + k0);
    acc0 = __builtin_amdgcn_wmma_f32_16x16x4_f32(false, a, false, b0, (short)0, acc0, false, false);
    acc1 = __builtin_amdgcn_wmma_f32_16x16x4_f32(false, a, false, b1, (short)0, acc1, false, false);
  }
  const int mb = (lane >> 4) << 3;
#pragma unroll
  for (int v = 0; v < 8; ++v) {
    const size_t o = (size_t)(m0 + mb + v) * D_MODEL + n0 + rr;
    out[o]      = acc0[v];
    out[o + 16] = acc1[v];
  }
}

// ---------------------------------------------------------------------------
extern "C" void kernel_launch(void* const* d_in, const int* in_sizes, int n_in,
                              void* d_out, int out_size, void* d_ws, size_t ws_size,
                              hipStream_t stream) {
  const float* hs    = (const float*)d_in[0];
  const float* in_w  = (const float*)d_in[1];
  const float* out_w = (const float*)d_in[2];
  const float* cw    = (const float*)d_in[3];
  const float* cb    = (const float*)d_in[4];
  const float* xw    = (const float*)d_in[5];
  const float* dw    = (const float*)d_in[6];
  const float* db    = (const float*)d_in[7];
  const float* Alog  = (const float*)d_in[8];
  const float* Dsk   = (const float*)d_in[9];
  float* out = (float*)d_out;
  float* ws  = (float*)d_ws;

  const size_t batched_floats =
      (size_t)SZ_XZ + SZ_TOT + 6ull * (SZ_U + SZ_XDBL + SZ_DELTA + SZ_TOT);
  const bool batched = ws_size >= batched_floats * sizeof(float);

  float* xz    = ws;
  float* total = xz + SZ_XZ;

  if (batched) {
    // all 6 branches concurrent (blockIdx.z); deterministic reduce, no atomics
    float* u6    = total + SZ_TOT;
    float* xd6   = u6  + 6ull * SZ_U;
    float* de6   = xd6 + 6ull * SZ_XDBL;
    float* y6    = de6 + 6ull * SZ_DELTA;

    k_in_proj<<<dim3(64, 128), 32, 0, stream>>>(hs, in_w, xz);
    k_conv_silu<<<dim3(8192, 1, 6), 256, 0, stream>>>(xz, cw, cb, u6, SZ_U, 0);
    k_xproj<<<dim3(3, 128, 6), 32, 0, stream>>>(u6, SZ_U, xw, xd6, SZ_XDBL, 0);
    k_dtproj<<<dim3(32, 128, 6), 32, 0, stream>>>(xd6, SZ_XDBL, dw, db, de6, SZ_DELTA, 0);
    k_scan<<<dim3(8, 1, 6), 128, 0, stream>>>(u6, SZ_U, de6, SZ_DELTA, xd6, SZ_XDBL,
                                              xz, Alog, Dsk, y6, SZ_TOT, 0, 0);
    k_reduce<<<8192, 256, 0, stream>>>(y6, total, SZ_TOT);
  } else {
    // branch buffers reused; branches serialized on the stream
    float* u     = total + SZ_TOT;
    float* xdbl  = u + SZ_U;
    float* delta = xdbl + SZ_XDBL;

    k_zero<<<8192, 256, 0, stream>>>(total, SZ_TOT);
    k_in_proj<<<dim3(64, 128), 32, 0, stream>>>(hs, in_w, xz);
    for (int br = 0; br < 6; ++br) {
      k_conv_silu<<<dim3(8192, 1, 1), 256, 0, stream>>>(xz, cw, cb, u, 0, br);
      k_xproj<<<dim3(3, 128, 1), 32, 0, stream>>>(u, 0, xw, xdbl, 0, br);
      k_dtproj<<<dim3(32, 128, 1), 32, 0, stream>>>(xdbl, 0, dw, db, delta, 0, br);
      k_scan<<<dim3(8, 1, 1), 128, 0, stream>>>(u, 0, delta, 0, xdbl, 0,
                                                xz, Alog, Dsk, total, 0, br, 1);
    }
  }
  k_out_proj<<<dim3(256, 8), 32, 0, stream>>>(total, out_w, out);
}